// NormalizingFlow_57904749085325
// MI455X (gfx1250) — compile-verified
//
#include <hip/hip_runtime.h>

// Normalizing planar flows: B=65536 rows, LATENT=40, NUM_FLOWS=30.
// HBM-bound streaming kernel (~650 MB read once -> ~28 us floor at 23.3 TB/s).
// - One lane per row; z + u double-buffer in VGPRs; w consumed on the fly.
// - gfx1250 global_prefetch_b8 stream 6 flows ahead (guaranteed line coverage).
// - Per-block b tile (contiguous 30 KB) staged via global_load_async_to_lds_b128
//   (ASYNCcnt) and read back from LDS: removes all strided b global traffic.

typedef float f4 __attribute__((ext_vector_type(4)));

#define LATENT 40
#define NF     30
#define NV     (LATENT / 4)   // 10 float4 chunks per length-40 vector
#define PFD    6              // prefetch distance in flows (~960B ahead/array)
#define TPB    256            // threads per block

// Branch-free fast tanh: 1 - 2/(exp(2x)+1).  exp overflow->inf => tanh->+1,
// underflow->0 => tanh->-1, x=0 => 0.  All-VALU, no branches.
static __device__ __forceinline__ float fast_tanh(float x) {
    const float e = __expf(2.0f * x);
    return 1.0f - __fdividef(2.0f, e + 1.0f);
}

__global__ __launch_bounds__(TPB) void nf_kernel(
    const float* __restrict__ zin,
    const float* __restrict__ w,
    const float* __restrict__ u,
    const float* __restrict__ bb,
    float* __restrict__ zout,
    float* __restrict__ ladj,
    int B)
{
    __shared__ float sb[TPB * NF];          // [tid][t], 30720 B, contiguous tile

    const int tid  = threadIdx.x;
    const int row0 = blockIdx.x * TPB;
    const int row  = row0 + tid;

    // ---- Stage the block's b tile: one contiguous 30720B span -> LDS ------
    // 1920 x 16B chunks, coalesced across lanes, tracked by ASYNCcnt.
    {
        const char* bsrc = (const char*)(bb + (size_t)row0 * NF);
        #pragma unroll
        for (int r = 0; r < (TPB * NF) / (4 * TPB) + 1; ++r) {   // 8 rounds
            const int c = r * TPB + tid;                          // chunk idx
            if (c < (TPB * NF) / 4) {
                const unsigned lds_off =
                    (unsigned)(size_t)(&sb[c * 4]);   // low 32 bits = LDS addr
                const char* g = bsrc + (size_t)c * 16;
                asm volatile("global_load_async_to_lds_b128 %0, %1, off"
                             :: "v"(lds_off), "v"(g)
                             : "memory");
            }
        }
        asm volatile("s_wait_asynccnt 0" ::: "memory");
    }
    __syncthreads();   // all threads reach here (no early return before this)

    if (row < B) {
        const f4* __restrict__ wp = (const f4*)(w + (size_t)row * (NF * LATENT));
        const f4* __restrict__ up = (const f4*)(u + (size_t)row * (NF * LATENT));

        // z lives in registers for the whole chain (40 VGPRs).
        f4 z[NV];
        {
            const f4* zp = (const f4*)(zin + (size_t)row * LATENT);
            #pragma unroll
            for (int i = 0; i < NV; ++i) z[i] = zp[i];
        }

        // Register double-buffer for u_t; w_t is consumed on the fly.
        f4 ucur[NV], unxt[NV];
        #pragma unroll
        for (int i = 0; i < NV; ++i)
            ucur[i] = __builtin_nontemporal_load(&up[i]);

        // Warm the prefetch pipeline for the first PFD flows.
        #pragma unroll
        for (int t = 1; t < PFD; ++t) {
            __builtin_prefetch((const char*)&wp[(size_t)t * NV], 0, 0);
            __builtin_prefetch((const char*)&wp[(size_t)t * NV] + 128, 0, 0);
            __builtin_prefetch((const char*)&up[(size_t)t * NV], 0, 0);
            __builtin_prefetch((const char*)&up[(size_t)t * NV] + 128, 0, 0);
        }

        float s = 0.0f;

        #pragma unroll 2   // turns the ucur<-unxt swap into register renaming
        for (int t = 0; t < NF; ++t) {
            // Register-prefetch next flow's u.
            const int tn = (t + 1 < NF) ? (t + 1) : t;
            #pragma unroll
            for (int i = 0; i < NV; ++i)
                unxt[i] = __builtin_nontemporal_load(&up[(size_t)tn * NV + i]);

            // global_prefetch_b8 PFD flows ahead; two 128B lines per array
            // per flow so every line of the 160B/flow stream gets covered.
            const int tp = (t + PFD < NF) ? (t + PFD) : (NF - 1);
            const char* wpc = (const char*)&wp[(size_t)tp * NV];
            const char* upc = (const char*)&up[(size_t)tp * NV];
            __builtin_prefetch(wpc, 0, 0);
            __builtin_prefetch(wpc + 128, 0, 0);
            __builtin_prefetch(upc, 0, 0);
            __builtin_prefetch(upc + 128, 0, 0);

            const float bt = sb[tid * NF + t];   // ds_load_b32 from staged tile

            // Stream w_t once, accumulating both dots (w.z and w.u).
            float dwz = 0.0f, dwu = 0.0f;
            #pragma unroll
            for (int i = 0; i < NV; ++i) {
                const f4 w4 = __builtin_nontemporal_load(&wp[(size_t)t * NV + i]);
                dwz = fmaf(w4.x, z[i].x, dwz);
                dwz = fmaf(w4.y, z[i].y, dwz);
                dwz = fmaf(w4.z, z[i].z, dwz);
                dwz = fmaf(w4.w, z[i].w, dwz);
                dwu = fmaf(w4.x, ucur[i].x, dwu);
                dwu = fmaf(w4.y, ucur[i].y, dwu);
                dwu = fmaf(w4.z, ucur[i].z, dwu);
                dwu = fmaf(w4.w, ucur[i].w, dwu);
            }

            const float tr = fast_tanh(dwz + bt);

            // z <- z + u_t * tr
            #pragma unroll
            for (int i = 0; i < NV; ++i) {
                z[i].x = fmaf(ucur[i].x, tr, z[i].x);
                z[i].y = fmaf(ucur[i].y, tr, z[i].y);
                z[i].z = fmaf(ucur[i].z, tr, z[i].z);
                z[i].w = fmaf(ucur[i].w, tr, z[i].w);
            }

            // ladj += log|1 + (1 - tr^2) * (w.u)|
            const float om = 1.0f - tr * tr;
            s += __logf(__builtin_fabsf(fmaf(om, dwu, 1.0f)));

            #pragma unroll
            for (int i = 0; i < NV; ++i) ucur[i] = unxt[i];
        }

        f4* zo = (f4*)(zout + (size_t)row * LATENT);
        #pragma unroll
        for (int i = 0; i < NV; ++i) zo[i] = z[i];
        ladj[row] = s;
    }
}

extern "C" void kernel_launch(void* const* d_in, const int* in_sizes, int n_in,
                              void* d_out, int out_size, void* d_ws, size_t ws_size,
                              hipStream_t stream) {
    const float* zin = (const float*)d_in[0];   // [B, 40]
    const float* w   = (const float*)d_in[1];   // [B, 30*40]
    const float* u   = (const float*)d_in[2];   // [B, 30*40]
    const float* b   = (const float*)d_in[3];   // [B, 30]

    const int B = in_sizes[0] / LATENT;

    float* zout = (float*)d_out;                        // [B, 40]
    float* ladj = (float*)d_out + (size_t)B * LATENT;   // [B, 1]

    dim3 block(TPB);
    dim3 grid((B + TPB - 1) / TPB);
    nf_kernel<<<grid, block, 0, stream>>>(zin, w, u, b, zout, ladj, B);
}